// GCNLayer_74079595921837
// MI455X (gfx1250) — compile-verified
//
#include <hip/hip_runtime.h>
#include <stdint.h>

// ---------------------------------------------------------------------------
// SpMM scatter: out[rows[e]] += vals[e] * embeds[cols[e]]   (E=1.6M, D=48)
//
// MI455X reasoning:
//  * Working set ~64MB < 192MB L2 -> after first replay the kernel is bound
//    by L2 request throughput, dominated by 76.8M global_atomic_add_f32.
//  * Streaming edge triples are double-buffered into LDS with gfx1250 async
//    b128 loads (ASYNCcnt / s_wait_asynccnt) -> 48 requests per 256-edge tile.
//  * Embeds gathered as aligned float4 (b128), 12 lanes per edge.
//  * WMMA inapplicable: 154 MFLOP total; no matmul structure without a sort.
// ---------------------------------------------------------------------------

#define D_FEAT   48
#define VEC_G    12            // 48 floats = 12 float4 groups per edge
#define TILE     256           // edges staged per LDS buffer
#define TPB      256           // threads per block (8 wave32)
#define TILES_PER_BLOCK 5

__device__ __forceinline__ void fatomic_add(float* p, float v) {
    // global_atomic_add_f32 (no return) on gfx1250.
    unsafeAtomicAdd(p, v);
}

__device__ __forceinline__ void async_copy_b128(uint32_t lds_addr, const void* gaddr) {
    asm volatile("global_load_async_to_lds_b128 %0, %1, off"
                 :: "v"(lds_addr), "v"(gaddr) : "memory");
}
__device__ __forceinline__ void async_copy_b32(uint32_t lds_addr, const void* gaddr) {
    asm volatile("global_load_async_to_lds_b32 %0, %1, off"
                 :: "v"(lds_addr), "v"(gaddr) : "memory");
}

__global__ __launch_bounds__(TPB)
void spmm_scatter_kernel(const int* __restrict__ rows,
                         const int* __restrict__ cols,
                         const float* __restrict__ vals,
                         const float* __restrict__ embeds,
                         float* __restrict__ out,
                         int nEdges, int nTiles)
{
    __shared__ alignas(16) int   s_rows[2][TILE];
    __shared__ alignas(16) int   s_cols[2][TILE];
    __shared__ alignas(16) float s_vals[2][TILE];

    const int tid      = threadIdx.x;
    const int tileBase = blockIdx.x * TILES_PER_BLOCK;
    if (tileBase >= nTiles) return;
    const int tileEnd  = (tileBase + TILES_PER_BLOCK < nTiles)
                       ? (tileBase + TILES_PER_BLOCK) : nTiles;

    // Stage one tile of (row, col, val) into LDS buffer `buf`.
    // Full tile: lanes 0..191 move 16B each via async b128 (1 op on waves 0-5,
    // 0 ops on waves 6-7 -> per-wave ASYNCcnt uniform per tile).
    auto stage = [&](int tile, int buf) {
        const int eBase = tile * TILE;
        if (eBase + TILE <= nEdges) {
            if (tid < 192) {
                const int arr = tid >> 6;    // wave-uniform: 0=rows 1=cols 2=vals
                const int l   = (tid & 63) * 4;
                const void* g;
                uint32_t lds;
                if (arr == 0) {
                    g = rows + eBase + l;  lds = (uint32_t)(uintptr_t)&s_rows[buf][l];
                } else if (arr == 1) {
                    g = cols + eBase + l;  lds = (uint32_t)(uintptr_t)&s_cols[buf][l];
                } else {
                    g = vals + eBase + l;  lds = (uint32_t)(uintptr_t)&s_vals[buf][l];
                }
                async_copy_b128(lds, g);
            }
        } else {                              // partial tail tile (robustness)
            const int e = eBase + tid;
            if (e < nEdges) {
                async_copy_b32((uint32_t)(uintptr_t)&s_rows[buf][tid], rows + e);
                async_copy_b32((uint32_t)(uintptr_t)&s_cols[buf][tid], cols + e);
                async_copy_b32((uint32_t)(uintptr_t)&s_vals[buf][tid], vals + e);
            }
        }
    };

    int buf = 0;
    stage(tileBase, 0);

    for (int t = tileBase; t < tileEnd; ++t, buf ^= 1) {
        const bool haveNext = (t + 1) < tileEnd;
        if (haveNext) {
            stage(t + 1, buf ^ 1);
            // Async loads retire in order per wave: <=1 outstanding means the
            // current tile's op is done (next tile's op may still be in
            // flight). Tail-tile staging issues more ops -> over-wait: safe.
            asm volatile("s_wait_asynccnt 0x1" ::: "memory");
        } else {
            asm volatile("s_wait_asynccnt 0x0" ::: "memory");
        }
        __syncthreads();   // all waves' LDS writes for `buf` visible

        const int eBase = t * TILE;
        int nE = nEdges - eBase;
        if (nE > TILE) nE = TILE;

        const int*   br = s_rows[buf];
        const int*   bc = s_cols[buf];
        const float* bv = s_vals[buf];

        if (nE == TILE) {
            // Constant trip count: 12 iterations per thread.
            #pragma unroll 4
            for (int k = tid; k < TILE * VEC_G; k += TPB) {
                const int el = k / VEC_G;
                const int g  = k - el * VEC_G;
                const int   row = br[el];    // same-address DS broadcast
                const int   col = bc[el];
                const float val = bv[el];
                const float4 x =
                    *((const float4*)(embeds + (size_t)col * D_FEAT) + g);
                float* dst = out + (size_t)row * D_FEAT + g * 4;
                fatomic_add(dst + 0, val * x.x);
                fatomic_add(dst + 1, val * x.y);
                fatomic_add(dst + 2, val * x.z);
                fatomic_add(dst + 3, val * x.w);
            }
        } else {
            for (int k = tid; k < nE * VEC_G; k += TPB) {
                const int el = k / VEC_G;
                const int g  = k - el * VEC_G;
                const int   row = br[el];
                const int   col = bc[el];
                const float val = bv[el];
                const float4 x =
                    *((const float4*)(embeds + (size_t)col * D_FEAT) + g);
                float* dst = out + (size_t)row * D_FEAT + g * 4;
                fatomic_add(dst + 0, val * x.x);
                fatomic_add(dst + 1, val * x.y);
                fatomic_add(dst + 2, val * x.z);
                fatomic_add(dst + 3, val * x.w);
            }
        }
        __syncthreads();   // done reading `buf` before it is restaged
    }
}

extern "C" void kernel_launch(void* const* d_in, const int* in_sizes, int n_in,
                              void* d_out, int out_size, void* d_ws, size_t ws_size,
                              hipStream_t stream) {
    const int*   rows   = (const int*)  d_in[0];
    const int*   cols   = (const int*)  d_in[1];
    const float* vals   = (const float*)d_in[2];
    const float* embeds = (const float*)d_in[3];
    float*       out    = (float*)d_out;

    const int nEdges = in_sizes[0];
    const int nTiles = (nEdges + TILE - 1) / TILE;
    const int grid   = (nTiles + TILES_PER_BLOCK - 1) / TILES_PER_BLOCK;

    // Scatter accumulates into d_out: zero it every call (graph-capturable).
    hipMemsetAsync(d_out, 0, (size_t)out_size * sizeof(float), stream);

    spmm_scatter_kernel<<<grid, TPB, 0, stream>>>(rows, cols, vals, embeds,
                                                  out, nEdges, nTiles);
}